// PointNet_Plus_1_55929064129271
// MI455X (gfx1250) — compile-verified
//
#include <hip/hip_runtime.h>
#include <hip/hip_bf16.h>

#define EPSF 1e-5f

typedef __attribute__((ext_vector_type(2))) float v2f;
typedef __attribute__((ext_vector_type(8))) float v8f;

struct Layer {
  const float* W; const float* b; const float* g; const float* be;
  const float* mu; const float* va;
};

// ---- WMMA helpers: V_WMMA_F32_16X16X4_F32 (native fp32 matrix op on CDNA5) ----
__device__ __forceinline__ v8f wmma4(v2f a, v2f b, v8f c) {
  return __builtin_amdgcn_wmma_f32_16x16x4_f32(
      /*neg_a=*/false, a, /*neg_b=*/false, b,
      /*c_mod=*/(short)0, c, /*reuse_a=*/false, /*reuse_b=*/false);
}

// A fragment (16x4, row-major W with leading dim ldw), zero-guard k >= kmax.
// ISA layout: lanes 0-15 hold K=0,1 ; lanes 16-31 hold K=2,3 ; M = lane%16.
__device__ __forceinline__ v2f load_a(const float* W, int ldw, int m0, int k0,
                                      int kmax, int mr, int half) {
  int k = k0 + 2 * half;
  const float* row = W + (m0 + mr) * ldw;
  v2f a;
  a.x = (k     < kmax) ? row[k]     : 0.f;
  a.y = (k + 1 < kmax) ? row[k + 1] : 0.f;
  return a;
}

// B fragment (4x16, row-major act with leading dim ldb), col = n0 + lane%16.
// Mirrors A: lanes 0-15 -> K=0,1 ; lanes 16-31 -> K=2,3.
__device__ __forceinline__ v2f load_b_row(const float* act, int ldb, int k0,
                                          int col, int half) {
  int k = k0 + 2 * half;
  v2f b;
  b.x = act[k * ldb + col];
  b.y = act[(k + 1) * ldb + col];
  return b;
}

__device__ __forceinline__ v2f load_b_row_guard(const float* act, int ldb, int k0,
                                                int col, int half, int kmax) {
  int k = k0 + 2 * half;
  v2f b;
  b.x = (k     < kmax) ? act[k * ldb + col]       : 0.f;
  b.y = (k + 1 < kmax) ? act[(k + 1) * ldb + col] : 0.f;
  return b;
}

// C/D layout: VGPR r, lane l -> row m0 + r + 8*(l/16), col n0 + l%16.
__device__ __forceinline__ void epilogue_lds(float* dst, int ldd, int m0, int n0,
                                             v8f acc, Layer L, int half, int nr) {
#pragma unroll
  for (int r = 0; r < 8; ++r) {
    int c = m0 + r + 8 * half;
    float s = L.g[c] * rsqrtf(L.va[c] + EPSF);
    float v = (acc[r] + L.b[c] - L.mu[c]) * s + L.be[c];
    dst[c * ldd + n0 + nr] = fmaxf(v, 0.f);
  }
}

// =================== Stage 1: fused b1 MLP (3->64->64->128) + max over k=80 ==
__global__ __launch_bounds__(256) void b1_fused_kernel(
    const float* __restrict__ x, Layer L1, Layer L2, Layer L3,
    float* __restrict__ out1) {
  extern __shared__ float smem[];
  float* A0 = smem;          // 4  x 80 (row 3 zeroed)
  float* A1 = smem + 320;    // 64 x 80
  float* A2 = smem + 5440;   // 64 x 80
  float* A3 = smem + 10560;  // 128 x 80   (total 20800 floats = 83,200 B)
  const int bidx = blockIdx.x;
  const int b = bidx >> 9, n = bidx & 511;
  const int tid = threadIdx.x;
  const int wave = tid >> 5, lane = tid & 31;
  const int half = lane >> 4, nr = lane & 15, mr = nr;

  for (int t = tid; t < 320; t += 256) {
    int c = t / 80, j = t % 80;
    A0[t] = (c < 3) ? x[((b * 3 + c) * 512 + n) * 80 + j] : 0.f;
  }
  __syncthreads();

  // layer1: 64x80, K=3 (padded to 4)
  for (int t = wave; t < 20; t += 8) {
    int mt = t / 5, nt = t % 5;
    v8f acc = {};
    acc = wmma4(load_a(L1.W, 3, mt * 16, 0, 3, mr, half),
                load_b_row(A0, 80, 0, nt * 16 + nr, half), acc);
    epilogue_lds(A1, 80, mt * 16, nt * 16, acc, L1, half, nr);
  }
  __syncthreads();

  // layer2: 64x80, K=64
  for (int t = wave; t < 20; t += 8) {
    int mt = t / 5, nt = t % 5;
    v8f acc = {};
    for (int k0 = 0; k0 < 64; k0 += 4)
      acc = wmma4(load_a(L2.W, 64, mt * 16, k0, 64, mr, half),
                  load_b_row(A1, 80, k0, nt * 16 + nr, half), acc);
    epilogue_lds(A2, 80, mt * 16, nt * 16, acc, L2, half, nr);
  }
  __syncthreads();

  // layer3: 128x80, K=64
  for (int t = wave; t < 40; t += 8) {
    int mt = t / 5, nt = t % 5;
    v8f acc = {};
    for (int k0 = 0; k0 < 64; k0 += 4)
      acc = wmma4(load_a(L3.W, 64, mt * 16, k0, 64, mr, half),
                  load_b_row(A2, 80, k0, nt * 16 + nr, half), acc);
    epilogue_lds(A3, 80, mt * 16, nt * 16, acc, L3, half, nr);
  }
  __syncthreads();

  // max over k=80
  for (int c = tid; c < 128; c += 256) {
    const float* row = A3 + c * 80;
    float mx = row[0];
    for (int j = 1; j < 80; ++j) mx = fmaxf(mx, row[j]);
    out1[(b * 512 + n) * 128 + c] = mx;
  }
}

// =================== Stage 2: KNN via bitonic sort of 512 distances =========
__global__ __launch_bounds__(256) void knn_kernel(const float* __restrict__ y,
                                                  int* __restrict__ idxOut) {
  __shared__ float dist[512];
  __shared__ int ind[512];
  __shared__ float cx[3];
  const int bidx = blockIdx.x;
  const int b = bidx >> 7, m = bidx & 127;
  const int tid = threadIdx.x;
  if (tid < 3) cx[tid] = y[(b * 3 + tid) * 512 + m];
  __syncthreads();
  for (int n = tid; n < 512; n += 256) {
    float dx = y[(b * 3 + 0) * 512 + n] - cx[0];
    float dy = y[(b * 3 + 1) * 512 + n] - cx[1];
    float dz = y[(b * 3 + 2) * 512 + n] - cx[2];
    dist[n] = dx * dx + dy * dy + dz * dz;
    ind[n] = n;
  }
  __syncthreads();
  // ascending bitonic sort on (dist, ind)
  for (int ks = 2; ks <= 512; ks <<= 1) {
    for (int jj = ks >> 1; jj > 0; jj >>= 1) {
      for (int i = tid; i < 512; i += 256) {
        int l = i ^ jj;
        if (l > i) {
          bool up = ((i & ks) == 0);
          float di = dist[i], dl = dist[l];
          int ii = ind[i], il = ind[l];
          bool sw = up ? (dl < di || (dl == di && il < ii))
                       : (di < dl || (di == dl && ii < il));
          if (sw) { dist[i] = dl; dist[l] = di; ind[i] = il; ind[l] = ii; }
        }
      }
      __syncthreads();
    }
  }
  if (tid < 64) {
    int sel = (dist[tid] > 0.15f) ? m : ind[tid];  // RADIUS = 0.25 - 0.1
    idxOut[(b * 128 + m) * 64 + tid] = sel;
  }
}

// ====== Stage 3: fused b2 MLP (131->128->128->256) + max over 64 neighbors ==
__global__ __launch_bounds__(256) void b2_fused_kernel(
    const float* __restrict__ y, const float* __restrict__ feat,
    const int* __restrict__ idx, Layer L1, Layer L2, Layer L3,
    float* __restrict__ in3) {
  extern __shared__ float smem[];
  float* IN0 = smem;          // 132 x 64 (rows 131 zeroed)
  float* A1 = smem + 8448;    // 128 x 64
  float* A2 = smem + 16640;   // 128 x 64
  float* A3 = smem + 24832;   // 256 x 64   (total 41216 floats = 164,864 B)
  __shared__ int nbr[64];
  const int bidx = blockIdx.x;
  const int b = bidx >> 7, m = bidx & 127;
  const int tid = threadIdx.x;
  const int wave = tid >> 5, lane = tid & 31;
  const int half = lane >> 4, nr = lane & 15, mr = nr;

  if (tid < 64) nbr[tid] = idx[(b * 128 + m) * 64 + tid];
  __syncthreads();

  for (int t = tid; t < 132 * 64; t += 256) {
    int c = t >> 6, j = t & 63;
    float v = 0.f;
    if (c < 131) {
      int nj = nbr[j];
      if (c < 3) v = y[(b * 3 + c) * 512 + nj] - y[(b * 3 + c) * 512 + m];
      else       v = feat[(b * 512 + nj) * 128 + (c - 3)];
    }
    IN0[t] = v;
  }
  __syncthreads();

  // layer1: 128x64, K=131 (pad 132)
  for (int t = wave; t < 32; t += 8) {
    int mt = t >> 2, nt = t & 3;
    v8f acc = {};
    for (int k0 = 0; k0 < 132; k0 += 4)
      acc = wmma4(load_a(L1.W, 131, mt * 16, k0, 131, mr, half),
                  load_b_row(IN0, 64, k0, nt * 16 + nr, half), acc);
    epilogue_lds(A1, 64, mt * 16, nt * 16, acc, L1, half, nr);
  }
  __syncthreads();

  // layer2: 128x64, K=128
  for (int t = wave; t < 32; t += 8) {
    int mt = t >> 2, nt = t & 3;
    v8f acc = {};
    for (int k0 = 0; k0 < 128; k0 += 4)
      acc = wmma4(load_a(L2.W, 128, mt * 16, k0, 128, mr, half),
                  load_b_row(A1, 64, k0, nt * 16 + nr, half), acc);
    epilogue_lds(A2, 64, mt * 16, nt * 16, acc, L2, half, nr);
  }
  __syncthreads();

  // layer3: 256x64, K=128
  for (int t = wave; t < 64; t += 8) {
    int mt = t >> 2, nt = t & 3;
    v8f acc = {};
    for (int k0 = 0; k0 < 128; k0 += 4)
      acc = wmma4(load_a(L3.W, 128, mt * 16, k0, 128, mr, half),
                  load_b_row(A2, 64, k0, nt * 16 + nr, half), acc);
    epilogue_lds(A3, 64, mt * 16, nt * 16, acc, L3, half, nr);
  }
  __syncthreads();

  // max over 64 neighbors; emit b3 input column [260 x 2048] (row 259 = pad)
  const int pos = b * 128 + m;
  for (int c = tid; c < 256; c += 256) {
    const float* row = A3 + c * 64;
    float mx = row[0];
    for (int j = 1; j < 64; ++j) mx = fmaxf(mx, row[j]);
    in3[(3 + c) * 2048 + pos] = mx;
  }
  if (tid < 3) in3[tid * 2048 + pos] = y[(b * 3 + tid) * 512 + m];
  if (tid == 3) in3[259 * 2048 + pos] = 0.f;
}

// =================== Stage 4: b3 GEMM layers over all 2048 positions ========
__global__ __launch_bounds__(256) void gemm_bn_relu_kernel(
    Layer L, const float* __restrict__ act, float* __restrict__ out,
    int cin, int krows) {
  const int npos = 2048;
  const int m0b = blockIdx.x * 64;
  const int n0b = blockIdx.y * 64;
  const int tid = threadIdx.x;
  const int wave = tid >> 5, lane = tid & 31;
  const int half = lane >> 4, nr = lane & 15, mr = nr;
  const int ksteps = (cin + 3) >> 2;
  for (int t = wave; t < 16; t += 8) {
    int mt = t >> 2, nt = t & 3;
    int m0 = m0b + mt * 16, n0 = n0b + nt * 16;
    v8f acc = {};
    for (int ks = 0; ks < ksteps; ++ks)
      acc = wmma4(load_a(L.W, cin, m0, ks * 4, cin, mr, half),
                  load_b_row_guard(act, npos, ks * 4, n0 + nr, half, krows), acc);
#pragma unroll
    for (int r = 0; r < 8; ++r) {
      int c = m0 + r + 8 * half;
      float s = L.g[c] * rsqrtf(L.va[c] + EPSF);
      float v = (acc[r] + L.b[c] - L.mu[c]) * s + L.be[c];
      out[c * npos + n0 + nr] = fmaxf(v, 0.f);
    }
  }
}

// =================== Stage 5: global max-pool + fc1 + fc2 ===================
__global__ __launch_bounds__(256) void head_kernel(
    const float* __restrict__ act /*1024 x 2048*/, Layer F1,
    const float* __restrict__ fc2W, const float* __restrict__ fc2B,
    float* __restrict__ outLogits) {
  __shared__ float feat[1024];
  __shared__ float h1[512];
  const int b = blockIdx.x;
  const int tid = threadIdx.x;
  for (int c = tid; c < 1024; c += 256) {
    const float* row = act + c * 2048 + b * 128;
    float mx = row[0];
    for (int m = 1; m < 128; ++m) mx = fmaxf(mx, row[m]);
    feat[c] = mx;
  }
  __syncthreads();
  for (int c = tid; c < 512; c += 256) {
    const float* w = F1.W + c * 1024;
    float acc = 0.f;
    for (int k = 0; k < 1024; ++k) acc += w[k] * feat[k];
    float s = F1.g[c] * rsqrtf(F1.va[c] + EPSF);
    float v = (acc + F1.b[c] - F1.mu[c]) * s + F1.be[c];
    h1[c] = fmaxf(v, 0.f);
  }
  __syncthreads();
  if (tid < 60) {
    const float* w = fc2W + tid * 512;
    float acc = 0.f;
    for (int k = 0; k < 512; ++k) acc += w[k] * h1[k];
    outLogits[b * 60 + tid] = acc + fc2B[tid];
  }
}

// ============================================================================
extern "C" void kernel_launch(void* const* d_in, const int* in_sizes, int n_in,
                              void* d_out, int out_size, void* d_ws, size_t ws_size,
                              hipStream_t stream) {
  (void)n_in; (void)out_size; (void)ws_size;
  // Flattening order hedge: insertion order (x, y, params...) vs sorted-pytree
  // order (params leaves..., x, y). Detect via x's unique size.
  const bool insOrder = (in_sizes[0] == 16 * 3 * 512 * 80);
  const float* x = (const float*)d_in[insOrder ? 0 : 62];
  const float* y = (const float*)d_in[insOrder ? 1 : 63];
  const int base = insOrder ? 2 : 0;
  auto mkLayer = [&](int i) -> Layer {
    Layer l;
    l.W = (const float*)d_in[i + 0];
    l.b = (const float*)d_in[i + 1];
    if (insOrder) {  // W, b, gamma, beta, mean, var
      l.g = (const float*)d_in[i + 2]; l.be = (const float*)d_in[i + 3];
    } else {         // sorted: W, b, beta, gamma, mean, var
      l.be = (const float*)d_in[i + 2]; l.g = (const float*)d_in[i + 3];
    }
    l.mu = (const float*)d_in[i + 4];
    l.va = (const float*)d_in[i + 5];
    return l;
  };
  Layer l11 = mkLayer(base + 0),  l12 = mkLayer(base + 6),  l13 = mkLayer(base + 12);
  Layer l21 = mkLayer(base + 18), l22 = mkLayer(base + 24), l23 = mkLayer(base + 30);
  Layer l31 = mkLayer(base + 36), l32 = mkLayer(base + 42), l33 = mkLayer(base + 48);
  Layer fc1 = mkLayer(base + 54);
  const float* fc2W = (const float*)d_in[base + 60];
  const float* fc2B = (const float*)d_in[base + 61];

  // workspace layout (floats)
  float* ws = (float*)d_ws;
  float* out1 = ws;                               // 16*512*128 = 1,048,576
  int*   idx  = (int*)(ws + 1048576);             // 16*128*64  =   131,072
  float* in3  = ws + 1048576 + 131072;            // 260*2048   =   532,480
  float* a3a  = in3 + 532480;                     // 256*2048   =   524,288
  float* a3b  = a3a + 524288;                     // 512*2048   = 1,048,576
  float* a3c  = a3b + 1048576;                    // 1024*2048  = 2,097,152

  b1_fused_kernel<<<16 * 512, 256, 20800 * sizeof(float), stream>>>(x, l11, l12, l13, out1);
  knn_kernel<<<16 * 128, 256, 0, stream>>>(y, idx);
  b2_fused_kernel<<<16 * 128, 256, 41216 * sizeof(float), stream>>>(y, out1, idx, l21, l22, l23, in3);
  gemm_bn_relu_kernel<<<dim3(4, 32),  256, 0, stream>>>(l31, in3, a3a, 259, 260);
  gemm_bn_relu_kernel<<<dim3(8, 32),  256, 0, stream>>>(l32, a3a, a3b, 256, 256);
  gemm_bn_relu_kernel<<<dim3(16, 32), 256, 0, stream>>>(l33, a3b, a3c, 512, 512);
  head_kernel<<<16, 256, 0, stream>>>(a3c, fc1, fc2W, fc2B, (float*)d_out);
}